// GPSConv_24927990186155
// MI455X (gfx1250) — compile-verified
//
#include <hip/hip_runtime.h>
#include <hip/hip_bf16.h>

// Problem constants (match reference)
#define NN 40000
#define LL 4
#define DD 128
#define EE 320000
#define LD2 (LL * DD)   // 512, row width of the [N, L*D] 2-d views

typedef float v2f __attribute__((ext_vector_type(2)));
typedef float v8f __attribute__((ext_vector_type(8)));

// ---------------------------------------------------------------------------
// CDNA5 fp32 WMMA: D(16x16,f32) = A(16x4,f32) * B(4x16,f32) + C
// A frag: lane m=lane&15 holds A[m][kboff+j], kboff = 2*(lane>>4)
// B frag: lane n=lane&15 holds B[kboff+j][n]
// C frag: lane holds rows M = j + 8*(lane>>4), col N = lane&15
// ---------------------------------------------------------------------------
__device__ __forceinline__ v8f wmma4(v2f a, v2f b, v8f c) {
  return __builtin_amdgcn_wmma_f32_16x16x4_f32(
      /*neg_a=*/false, a, /*neg_b=*/false, b,
      /*c_mod=*/(short)0, c, /*reuse_a=*/false, /*reuse_b=*/false);
}

__device__ __forceinline__ float silu_f(float v) {
  return v / (1.f + __expf(-v));
}

// Sum across the 16 lanes of a half-wave (lanes 0-15 or 16-31).
__device__ __forceinline__ float hsum16(float v) {
  v += __shfl_xor(v, 1, 32);
  v += __shfl_xor(v, 2, 32);
  v += __shfl_xor(v, 4, 32);
  v += __shfl_xor(v, 8, 32);
  return v;
}

// 16 rows x 128 cols tile: acc += A(16x128) @ W(128x128), W row-major.
template <class AF>
__device__ __forceinline__ void gemm128(AF loadA, const float* __restrict__ W,
                                        int lane, v8f acc[8]) {
  const int n = lane & 15;
  const int kboff = (lane >> 4) << 1;
  for (int kk = 0; kk < 32; ++kk) {
    const int kb = kk * 4 + kboff;
    v2f a = loadA(kb);
#pragma unroll
    for (int cb = 0; cb < 8; ++cb) {
      v2f b;
      b.x = W[(size_t)kb * DD + cb * 16 + n];
      b.y = W[(size_t)(kb + 1) * DD + cb * 16 + n];
      acc[cb] = wmma4(a, b, acc[cb]);
    }
  }
}

// 16 rows x 16 cols tile: acc += A(16x128) @ W(128x16), W row-major.
template <class AF>
__device__ __forceinline__ void gemm16(AF loadA, const float* __restrict__ W,
                                       int lane, v8f& acc) {
  const int n = lane & 15;
  const int kboff = (lane >> 4) << 1;
  for (int kk = 0; kk < 32; ++kk) {
    const int kb = kk * 4 + kboff;
    v2f a = loadA(kb);
    v2f b;
    b.x = W[(size_t)kb * 16 + n];
    b.y = W[(size_t)(kb + 1) * 16 + n];
    acc = wmma4(a, b, acc);
  }
}

// ---------------------------------------------------------------------------
// Kernel 1: zero workspace
// ---------------------------------------------------------------------------
__global__ void __launch_bounds__(256) zero_kernel(float4* p, long n4) {
  long i = (long)blockIdx.x * blockDim.x + threadIdx.x;
  if (i < n4) p[i] = make_float4(0.f, 0.f, 0.f, 0.f);
}

// ---------------------------------------------------------------------------
// Kernel 2: agg2d[rows[e], :] += vals[e] * x2d[cols[e], :]
// (x2d row n = concat over l of x[l,n,:], gathered directly from x)
// One wave per edge; lane covers 16 of the 512 row elements.
// ---------------------------------------------------------------------------
__global__ void __launch_bounds__(256) spmm_x_kernel(
    const int* __restrict__ rows, const int* __restrict__ cols,
    const float* __restrict__ vals, const float* __restrict__ x,
    float* __restrict__ agg2d) {
  const int wv = threadIdx.x >> 5, lane = threadIdx.x & 31;
  const long e = (long)blockIdx.x * 8 + wv;
  const int rdst = rows[e];
  const int csrc = cols[e];
  const float v = vals[e];
  float* dst = agg2d + (size_t)rdst * LD2;
#pragma unroll
  for (int i = 0; i < 16; ++i) {
    const int idx = lane + 32 * i;          // 0..511
    const int l = idx >> 7, d = idx & 127;  // layer, feature
    const float s = x[((size_t)l * NN + csrc) * DD + d];
    __hip_atomic_fetch_add(dst + idx, v * s, __ATOMIC_RELAXED,
                           __HIP_MEMORY_SCOPE_AGENT);
  }
}

// ---------------------------------------------------------------------------
// Kernel 3: dst2d[rows[e], :] += scale * vals[e] * src2d[cols[e], :]
// ---------------------------------------------------------------------------
__global__ void __launch_bounds__(256) spmm_2d_kernel(
    const int* __restrict__ rows, const int* __restrict__ cols,
    const float* __restrict__ vals, const float* __restrict__ src2d,
    float* __restrict__ dst2d, float scale) {
  const int wv = threadIdx.x >> 5, lane = threadIdx.x & 31;
  const long e = (long)blockIdx.x * 8 + wv;
  const int rdst = rows[e];
  const int csrc = cols[e];
  const float v = scale * vals[e];
  const float* src = src2d + (size_t)csrc * LD2;
  float* dst = dst2d + (size_t)rdst * LD2;
#pragma unroll
  for (int i = 0; i < 16; ++i) {
    const int idx = lane + 32 * i;
    __hip_atomic_fetch_add(dst + idx, v * src[idx], __ATOMIC_RELAXED,
                           __HIP_MEMORY_SCOPE_AGENT);
  }
}

// ---------------------------------------------------------------------------
// Kernel 4: retention weights
// weights[r] = mean_k( (x@K)*(x@Q) ) + same for xr1=0.5*agg and xr2=cur2
// One wave per 16-row tile; six K=128 -> KD=16 WMMA GEMMs per tile.
// ---------------------------------------------------------------------------
__global__ void __launch_bounds__(256) weights_kernel(
    const float* __restrict__ x, const float* __restrict__ agg2d,
    const float* __restrict__ cur2, const float* __restrict__ keyW,
    const float* __restrict__ queryW, float* __restrict__ wout) {
  const int wv = threadIdx.x >> 5, lane = threadIdx.x & 31;
  const long r0 = ((long)blockIdx.x * 8 + wv) * 16;
  const int m = lane & 15, half = lane >> 4;
  const long r = r0 + m;
  const int l = (int)(r / NN), nn = (int)(r % NN);
  const float* xrow = x + (size_t)r * DD;
  const float* arow = agg2d + (size_t)nn * LD2 + (size_t)l * DD;
  const float* crow = cur2 + (size_t)nn * LD2 + (size_t)l * DD;

  const v8f vzero = {0.f, 0.f, 0.f, 0.f, 0.f, 0.f, 0.f, 0.f};
  float wsum[8];
#pragma unroll
  for (int j = 0; j < 8; ++j) wsum[j] = 0.f;

  {  // source 0: x
    v8f kp = vzero, qp = vzero;
    auto la = [&](int kb) { v2f a; a.x = xrow[kb]; a.y = xrow[kb + 1]; return a; };
    gemm16(la, keyW, lane, kp);
    gemm16(la, queryW, lane, qp);
#pragma unroll
    for (int j = 0; j < 8; ++j) wsum[j] += kp[j] * qp[j];
  }
  {  // source 1: xr1 = 0.5 * agg
    v8f kp = vzero, qp = vzero;
    auto la = [&](int kb) { v2f a; a.x = 0.5f * arow[kb]; a.y = 0.5f * arow[kb + 1]; return a; };
    gemm16(la, keyW, lane, kp);
    gemm16(la, queryW, lane, qp);
#pragma unroll
    for (int j = 0; j < 8; ++j) wsum[j] += kp[j] * qp[j];
  }
  {  // source 2: xr2 = cur2 (0.25 folded in by spmm scale)
    v8f kp = vzero, qp = vzero;
    auto la = [&](int kb) { v2f a; a.x = crow[kb]; a.y = crow[kb + 1]; return a; };
    gemm16(la, keyW, lane, kp);
    gemm16(la, queryW, lane, qp);
#pragma unroll
    for (int j = 0; j < 8; ++j) wsum[j] += kp[j] * qp[j];
  }

  // Row sums across the 16 columns live in one half-wave -> shuffle reduce.
#pragma unroll
  for (int j = 0; j < 8; ++j) {
    const float tot = hsum16(wsum[j]) * (1.f / 16.f);
    if (m == 0) wout[r0 + half * 8 + j] = tot;  // row M = j + 8*half
  }
}

// ---------------------------------------------------------------------------
// Kernel 5: fused sage + attention + GPS fusion, one wave per 16x128 tile.
// ---------------------------------------------------------------------------
#define LSTR 132  // padded row stride (floats) to avoid 64-bank conflicts

__global__ void __launch_bounds__(128) fused_kernel(
    const float* __restrict__ x, const float* __restrict__ agg2d,
    const float* __restrict__ cur2, const float* __restrict__ wrow,
    const float* __restrict__ sageW, const float* __restrict__ sageB,
    const float* __restrict__ sageAggW, const float* __restrict__ valueW,
    const float* __restrict__ attnG, const float* __restrict__ attnB,
    const float* __restrict__ linW, const float* __restrict__ linB,
    const float* __restrict__ lnG, const float* __restrict__ lnB,
    float* __restrict__ out) {
  __shared__ float ldsbuf[4 * 16 * LSTR];
  const int wv = threadIdx.x >> 5, lane = threadIdx.x & 31;
  float* stage = ldsbuf + wv * 16 * LSTR;
  const long r0 = ((long)blockIdx.x * 4 + wv) * 16;
  const int m = lane & 15, half = lane >> 4;
  const long r = r0 + m;
  const int l = (int)(r / NN), nn = (int)(r % NN);
  const float* xrow = x + (size_t)r * DD;
  const float* arow = agg2d + (size_t)nn * LD2 + (size_t)l * DD;
  const float* crow = cur2 + (size_t)nn * LD2 + (size_t)l * DD;

  const v8f vzero = {0.f, 0.f, 0.f, 0.f, 0.f, 0.f, 0.f, 0.f};
  v8f acc[8];

  // ---- Stage 1: sage = silu(x@sageW + agg@sageAggW + sage_b) ----
#pragma unroll
  for (int cb = 0; cb < 8; ++cb) acc[cb] = vzero;
  gemm128([&](int kb) { v2f a; a.x = xrow[kb]; a.y = xrow[kb + 1]; return a; },
          sageW, lane, acc);
  gemm128([&](int kb) { v2f a; a.x = arow[kb]; a.y = arow[kb + 1]; return a; },
          sageAggW, lane, acc);
  v8f sage[8];
#pragma unroll
  for (int cb = 0; cb < 8; ++cb) {
    const float bb = sageB[cb * 16 + m];
#pragma unroll
    for (int j = 0; j < 8; ++j) sage[cb][j] = silu_f(acc[cb][j] + bb);
  }

  // ---- Stage 2: attn = LN( (x_acc @ valueW) * w[row] ) ----
  // x_acc = x + 0.5*agg + cur2
#pragma unroll
  for (int cb = 0; cb < 8; ++cb) acc[cb] = vzero;
  gemm128(
      [&](int kb) {
        v2f a;
        a.x = xrow[kb] + 0.5f * arow[kb] + crow[kb];
        a.y = xrow[kb + 1] + 0.5f * arow[kb + 1] + crow[kb + 1];
        return a;
      },
      valueW, lane, acc);

  float wr[8];
#pragma unroll
  for (int j = 0; j < 8; ++j) wr[j] = wrow[r0 + half * 8 + j];

  // Register-only layernorm stats: row M=j+8*half spans this half-wave.
  float mu[8], rs[8];
#pragma unroll
  for (int j = 0; j < 8; ++j) {
    float s = 0.f, q = 0.f;
#pragma unroll
    for (int cb = 0; cb < 8; ++cb) {
      const float t = acc[cb][j] * wr[j];
      s += t; q += t * t;
    }
    s = hsum16(s); q = hsum16(q);
    const float mm = s * (1.f / 128.f);
    const float vv = q * (1.f / 128.f) - mm * mm;
    mu[j] = mm;
    rs[j] = rsqrtf(vv + 1e-5f);
  }

  // h = sage + attn_out -> LDS in row-major for A-fragment reads.
#pragma unroll
  for (int cb = 0; cb < 8; ++cb) {
    const int col = cb * 16 + m;
    const float g = attnG[col], b = attnB[col];
#pragma unroll
    for (int j = 0; j < 8; ++j) {
      const float t = acc[cb][j] * wr[j];
      const float an = (t - mu[j]) * rs[j] * g + b;
      stage[(j + 8 * half) * LSTR + col] = sage[cb][j] + an;
    }
  }
  __syncthreads();

  // ---- Stage 3: y = silu(h@linW + lin_b); out = LN(y) ----
#pragma unroll
  for (int cb = 0; cb < 8; ++cb) acc[cb] = vzero;
  gemm128(
      [&](int kb) {
        v2f a;
        a.x = stage[m * LSTR + kb];
        a.y = stage[m * LSTR + kb + 1];
        return a;
      },
      linW, lane, acc);
#pragma unroll
  for (int cb = 0; cb < 8; ++cb) {
    const float bb = linB[cb * 16 + m];
#pragma unroll
    for (int j = 0; j < 8; ++j) acc[cb][j] = silu_f(acc[cb][j] + bb);
  }
#pragma unroll
  for (int j = 0; j < 8; ++j) {
    float s = 0.f, q = 0.f;
#pragma unroll
    for (int cb = 0; cb < 8; ++cb) { const float t = acc[cb][j]; s += t; q += t * t; }
    s = hsum16(s); q = hsum16(q);
    const float mm = s * (1.f / 128.f);
    const float vv = q * (1.f / 128.f) - mm * mm;
    mu[j] = mm;
    rs[j] = rsqrtf(vv + 1e-5f);
  }
#pragma unroll
  for (int cb = 0; cb < 8; ++cb) {
    const int col = cb * 16 + m;
    const float g = lnG[col], b = lnB[col];
#pragma unroll
    for (int j = 0; j < 8; ++j)
      out[(size_t)(r0 + j + 8 * half) * DD + col] =
          (acc[cb][j] - mu[j]) * rs[j] * g + b;
  }
}

// ---------------------------------------------------------------------------
extern "C" void kernel_launch(void* const* d_in, const int* in_sizes, int n_in,
                              void* d_out, int out_size, void* d_ws,
                              size_t ws_size, hipStream_t stream) {
  const float* x        = (const float*)d_in[0];
  const int*   erows    = (const int*)d_in[1];
  const int*   ecols    = (const int*)d_in[2];
  const float* evals    = (const float*)d_in[3];
  const float* sageW    = (const float*)d_in[4];
  const float* sageB    = (const float*)d_in[5];
  const float* sageAggW = (const float*)d_in[6];
  const float* keyW     = (const float*)d_in[7];
  const float* queryW   = (const float*)d_in[8];
  const float* valueW   = (const float*)d_in[9];
  const float* attnG    = (const float*)d_in[10];
  const float* attnB    = (const float*)d_in[11];
  const float* linW     = (const float*)d_in[12];
  const float* linB     = (const float*)d_in[13];
  const float* lnG      = (const float*)d_in[14];
  const float* lnB      = (const float*)d_in[15];

  // Workspace: agg2d [N,512], cur2 [N,512], wrow [L*N]
  float* agg2d = (float*)d_ws;
  float* cur2  = agg2d + (size_t)NN * LD2;
  float* wrow  = cur2 + (size_t)NN * LD2;

  {  // zero agg2d + cur2 (contiguous)
    const long n4 = (long)2 * NN * LD2 / 4;
    zero_kernel<<<(unsigned)((n4 + 255) / 256), 256, 0, stream>>>(
        (float4*)agg2d, n4);
  }
  // agg = spmm(x2d)  (cur1 = 0.5*agg algebraically)
  spmm_x_kernel<<<EE / 8, 256, 0, stream>>>(erows, ecols, evals, x, agg2d);
  // cur2 = 0.25 * spmm(agg)
  spmm_2d_kernel<<<EE / 8, 256, 0, stream>>>(erows, ecols, evals, agg2d, cur2,
                                             0.25f);
  // retention weights per row
  weights_kernel<<<(LL * NN) / (16 * 8), 256, 0, stream>>>(x, agg2d, cur2,
                                                           keyW, queryW, wrow);
  // fused sage + attention + fusion + layernorms
  fused_kernel<<<(LL * NN) / (16 * 4), 128, 0, stream>>>(
      x, agg2d, cur2, wrow, sageW, sageB, sageAggW, valueW, attnG, attnB, linW,
      linB, lnG, lnB, (float*)d_out);
}